// single_channel_interp_10685878633056
// MI455X (gfx1250) — compile-verified
//
#include <hip/hip_runtime.h>
#include <hip/hip_bf16.h>

typedef __attribute__((ext_vector_type(2))) float v2f;
typedef __attribute__((ext_vector_type(8))) float v8f;

#define B_DIM 32
#define D_DIM 64
#define T_DIM 256
#define R_DIM 96
#define HOURS 48.0f
#define NEGBIG (-1e30f)

// One block (192 threads = 6 waves) per (b,d) pair.
// Wave w handles the 16-wide reference-time tile r in [16w, 16w+16).
__global__ __launch_bounds__(192) void sci_wmma_kernel(const float* __restrict__ x,
                                                       const float* __restrict__ kern,
                                                       float* __restrict__ out) {
  // LDS: [0,256) values, [256,512) mask, [512,768) timestamps
  __shared__ __align__(16) float sdata[3 * T_DIM];
  __shared__ float st[6 * 16 * 4];   // per-r staged {Y1,S1,Y10,S10}
  __shared__ float stmx[6 * 16];     // per-r max logit

  const int pair = blockIdx.x;
  const int b = pair >> 6;
  const int d = pair & 63;
  const int tid = threadIdx.x;

  // Stage three contiguous 256-float rows: exactly 192 float4 loads, one per thread.
  {
    const int seg = tid / 64;                 // 0: vals, 1: mask, 2: ts
    const int off = tid & 63;
    const int chan = (seg == 0) ? d : ((seg == 1) ? (64 + d) : (128 + d));
    const float4* src = (const float4*)(x + (size_t)(b * 256 + chan) * T_DIM);
    ((float4*)sdata)[seg * 64 + off] = src[off];
  }
  __syncthreads();

  const float* xv = sdata;
  const float* mk = sdata + T_DIM;
  const float* ts = sdata + 2 * T_DIM;

  // alpha = softplus(kernel[d]), numerically stable
  const float kv = kern[d];
  const float alpha = fmaxf(kv, 0.0f) + log1pf(__expf(-fabsf(kv)));

  const int wave = tid >> 5;                  // 0..5 -> r tile
  const int lane = tid & 31;
  const int half = lane >> 4;                 // 0: K=0,1 / t%8<4 ; 1: K=2,3 / t%8>=4
  const int n = lane & 15;                    // N column (B/D) == M row (A)
  const int r0 = wave * 16;
  const float ref_r = (float)(r0 + n) * (HOURS / 95.0f);

  // ---- Phase 1: masked max of scale-1 logits (branch-free, float4 LDS loads) ----
  float mx = NEGBIG;
  for (int tb = 0; tb < T_DIM; tb += 8) {
    const int base = tb + half * 4;
    float4 t4 = *(const float4*)(ts + base);
    float4 m4 = *(const float4*)(mk + base);
    float df, l;
    df = t4.x - ref_r; l = -alpha * df * df; mx = fmaxf(mx, (m4.x > 0.f) ? l : NEGBIG);
    df = t4.y - ref_r; l = -alpha * df * df; mx = fmaxf(mx, (m4.y > 0.f) ? l : NEGBIG);
    df = t4.z - ref_r; l = -alpha * df * df; mx = fmaxf(mx, (m4.z > 0.f) ? l : NEGBIG);
    df = t4.w - ref_r; l = -alpha * df * df; mx = fmaxf(mx, (m4.w > 0.f) ? l : NEGBIG);
  }
  mx = fmaxf(mx, __shfl_xor(mx, 16, 32));
  // logits(scale=10) = 10*logits(scale=1) => max10 = 10*mx, e10 = e1^10.

  // ---- Phase 2: WMMA-accumulated reductions ----
  // Per 8-t chunk: 2 WMMA K-steps; lane's 4 K-slots are one contiguous float4.
  const bool is_c0 = (n == 0);
  const float c1sel = (n == 1) ? 1.0f : 0.0f;
  v8f c1 = {};
  v8f c10 = {};
  for (int tb = 0; tb < T_DIM; tb += 8) {
    const int base = tb + half * 4;
    float4 t4 = *(const float4*)(ts + base);
    float4 m4 = *(const float4*)(mk + base);
    float4 x4 = *(const float4*)(xv + base);

    // helper lambda-free expansion for the 4 exp-weights
    float e0, e1e, e2, e3;
    {
      float df, s;
      df = t4.x - ref_r; s = -alpha * df * df - mx; e0  = m4.x * __expf(s);
      df = t4.y - ref_r; s = -alpha * df * df - mx; e1e = m4.y * __expf(s);
      df = t4.z - ref_r; s = -alpha * df * df - mx; e2  = m4.z * __expf(s);
      df = t4.w - ref_r; s = -alpha * df * df - mx; e3  = m4.w * __expf(s);
    }
    float p2, p4, p8;
    p2 = e0 * e0;  p4 = p2 * p2; p8 = p4 * p4; const float f0 = p8 * p2;
    p2 = e1e * e1e; p4 = p2 * p2; p8 = p4 * p4; const float f1 = p8 * p2;
    p2 = e2 * e2;  p4 = p2 * p2; p8 = p4 * p4; const float f2 = p8 * p2;
    p2 = e3 * e3;  p4 = p2 * p2; p8 = p4 * p4; const float f3 = p8 * p2;

    // WMMA step s=0: elements 0,1
    v2f a1, a10, bf;
    a1.x = e0;  a1.y = e1e;
    a10.x = f0; a10.y = f1;
    bf.x = is_c0 ? x4.x : c1sel;
    bf.y = is_c0 ? x4.y : c1sel;
    c1  = __builtin_amdgcn_wmma_f32_16x16x4_f32(false, a1,  false, bf, (short)0, c1,
                                                false, false);
    c10 = __builtin_amdgcn_wmma_f32_16x16x4_f32(false, a10, false, bf, (short)0, c10,
                                                false, false);

    // WMMA step s=1: elements 2,3
    a1.x = e2;  a1.y = e3;
    a10.x = f2; a10.y = f3;
    bf.x = is_c0 ? x4.z : c1sel;
    bf.y = is_c0 ? x4.w : c1sel;
    c1  = __builtin_amdgcn_wmma_f32_16x16x4_f32(false, a1,  false, bf, (short)0, c1,
                                                false, false);
    c10 = __builtin_amdgcn_wmma_f32_16x16x4_f32(false, a10, false, bf, (short)0, c10,
                                                false, false);
  }

  // ---- Extraction: columns 0/1 of D hold (sum e*x, sum e); stage via LDS ----
  if (n < 2) {
#pragma unroll
    for (int j = 0; j < 8; ++j) {
      const int rl = (half << 3) + j;         // row M = j (low lanes) or j+8 (high lanes)
      st[(wave * 16 + rl) * 4 + n]     = c1[j];
      st[(wave * 16 + rl) * 4 + 2 + n] = c10[j];
    }
  }
  if (lane < 16) stmx[wave * 16 + n] = mx;
  __syncthreads();

  if (tid < R_DIM) {
    const int r = tid;                        // global reference index 0..95
    const float Y1  = st[r * 4 + 0];
    const float S1  = st[r * 4 + 1];
    const float Y10 = st[r * 4 + 2];
    const float S10 = st[r * 4 + 3];
    const float m   = stmx[r];
    const size_t obase = (size_t)b * (3 * D_DIM) * R_DIM;
    out[obase + (size_t)d * R_DIM + r]               = Y1 / S1;          // y (scale 1)
    out[obase + (size_t)(D_DIM + d) * R_DIM + r]     = m + __logf(S1);   // w = logsumexp
    out[obase + (size_t)(2 * D_DIM + d) * R_DIM + r] = Y10 / S10;        // y_trans
  }
}

extern "C" void kernel_launch(void* const* d_in, const int* in_sizes, int n_in,
                              void* d_out, int out_size, void* d_ws, size_t ws_size,
                              hipStream_t stream) {
  const float* x = (const float*)d_in[0];       // (32, 256, 256) f32
  const float* kern = (const float*)d_in[1];    // (64,) f32
  float* out = (float*)d_out;                   // (32, 192, 96) f32
  dim3 grid(B_DIM * D_DIM);
  dim3 block(192);
  sci_wmma_kernel<<<grid, block, 0, stream>>>(x, kern, out);
}